// DWT2d_69561290326631
// MI455X (gfx1250) — compile-verified
//
#include <hip/hip_runtime.h>
#include <hip/hip_bf16.h>

typedef __attribute__((ext_vector_type(2))) float v2f;
typedef __attribute__((ext_vector_type(8))) float v8f;

#define B_   16
#define C_   32
#define H_   512
#define W_   512
#define OH_  259
#define OW_  259
#define TLO  32          // 32x32 output tile per workgroup
#define XR   72          // LDS input rows: exactly the valid/needed range
#define XC   76          // LDS input cols: 19 x 16B chunks (aligned base, 2-col left slack)
#define NCHK 19          // 16B column chunks per row
#define NTW  9           // ceil(259/32) tiles per axis

// db4 filters, pre-flipped for cross-correlation (lo = dec_lo[::-1], hi[k] = (-1)^k dec_lo[k])
static __device__ const float LOF[8] = {
     0.23037781330885523f,  0.7148465705525415f,   0.6308807679295904f,  -0.02798376941698385f,
    -0.18703481171888114f,  0.030841381835986965f, 0.032883011666982945f,-0.010597401784997278f };
static __device__ const float HIF[8] = {
    -0.010597401784997278f,-0.032883011666982945f, 0.030841381835986965f, 0.18703481171888114f,
    -0.02798376941698385f, -0.6308807679295904f,   0.7148465705525415f,  -0.23037781330885523f };

__global__ __launch_bounds__(128)
void dwt2_db4_wmma(const float* __restrict__ x, float* __restrict__ out) {
    __shared__ float sX[XR][XC];          // input tile, global col = IWB + c  (conv col = c - 2)
    __shared__ float sMid[2][XR][TLO];    // horizontal pass result, per filter
    __shared__ float sFilt[2][80];        // padded Toeplitz lookup: coeff at [32..39], 0 elsewhere

    const int tid  = threadIdx.x;
    const int lane = tid & 31;
    const int wv   = tid >> 5;            // 0..3
    const int qi   = wv >> 1;             // quadrant row (0/1)
    const int qj   = wv & 1;              // quadrant col (0/1)
    const int lm   = lane & 15;           // M / N index within 16
    const int lh   = lane >> 4;           // half select (K split)

    const int tile = blockIdx.x;
    const int ch   = blockIdx.y;
    const int b    = blockIdx.z;
    const int OH0  = TLO * (tile / NTW);
    const int OW0  = TLO * (tile % NTW);
    const int IH0  = 2 * OH0 - 6;         // pad (6,7)
    const int IWB  = 2 * OW0 - 8;         // aligned (16B) column base; conv origin at LDS col 2

    // ---- build padded filter tables in LDS ----
    for (int i = tid; i < 2 * 80; i += 128) {
        int f = i / 80, p = i % 80;
        float v = 0.0f;
        if (p >= 32 && p < 40) v = f ? HIF[p - 32] : LOF[p - 32];
        sFilt[f][p] = v;
    }

    // ---- stage input tile into LDS via async global->LDS B128 (zero OOB) ----
    const float* chbase = x + (size_t)(b * C_ + ch) * (size_t)(H_ * W_);
    const bool interior = (IH0 >= 0) && (IH0 + XR <= H_) && (IWB >= 0) && (IWB + XC <= W_);
    if (interior) {
        for (int i = tid; i < XR * NCHK; i += 128) {
            const int r = i / NCHK, ck = i % NCHK;
            unsigned ldsoff = (unsigned)(uintptr_t)&sX[r][4 * ck];
            unsigned voff   = (unsigned)(((IH0 + r) * W_ + IWB + 4 * ck) * 4);
            asm volatile("global_load_async_to_lds_b128 %0, %1, %2"
                         :: "v"(ldsoff), "v"(voff), "s"(chbase) : "memory");
        }
    } else {
        for (int i = tid; i < XR * NCHK; i += 128) {
            const int r = i / NCHK, ck = i % NCHK;
            const int ih = IH0 + r;
            const int iwc = IWB + 4 * ck;     // 16B chunk fully in or fully out of [0,512)
            unsigned ldsoff = (unsigned)(uintptr_t)&sX[r][4 * ck];
            if ((unsigned)ih < (unsigned)H_ && (unsigned)iwc < (unsigned)W_) {
                unsigned voff = (unsigned)((ih * W_ + iwc) * 4);
                asm volatile("global_load_async_to_lds_b128 %0, %1, %2"
                             :: "v"(ldsoff), "v"(voff), "s"(chbase) : "memory");
            } else {
                float4 z = {0.0f, 0.0f, 0.0f, 0.0f};
                *(float4*)&sX[r][4 * ck] = z;     // padding: must be finite-zero
            }
        }
    }
    asm volatile("s_wait_asynccnt 0" ::: "memory");
    __syncthreads();

    // ---- horizontal pass: Mid(16x16) = X(16x40) * H_toep(40x16), K in 10 steps of 4 ----
    // row chunks {0,16,24}+32*qi cover mid rows [32qi, 32qi+39] without touching sX row > 71
    const int W0  = 32 * qj + 2;          // LDS col base of this quadrant's K range (+2 align slack)
    const int OWq = 16 * qj;              // mid col base
    const int RCH[3] = {0, 16, 24};
    for (int f = 0; f < 2; ++f) {
        #pragma unroll
        for (int t = 0; t < 3; ++t) {
            const int R = 32 * qi + RCH[t];
            v8f acc = {};
            #pragma unroll
            for (int s = 0; s < 10; ++s) {
                v2f a;                    // A[m][4s + K], K = 2*lh (+1)
                const int ar = R + lm;
                const int ac = W0 + 4 * s + 2 * lh;
                a.x = sX[ar][ac];
                a.y = sX[ar][ac + 1];
                v2f bv;                   // B[K][n] = fh[(4s+K) - 2n], K = j + 2*lh
                const int k0 = 4 * s + 2 * lh - 2 * lm;
                bv.x = sFilt[f][32 + k0];
                bv.y = sFilt[f][32 + k0 + 1];
                acc = __builtin_amdgcn_wmma_f32_16x16x4_f32(
                        false, a, false, bv, (short)0, acc, false, false);
            }
            #pragma unroll
            for (int v = 0; v < 8; ++v)   // D vgpr v -> row v (lanes<16) / v+8 (lanes>=16)
                sMid[f][R + v + 8 * lh][OWq + lm] = acc[v];
        }
    }

    // ---- vertical pass: Out(16x16) = G_toep(16x40) * Mid(40x16), 4 subbands ----
    const int RB = 32 * qi;
    float* obase = out + (size_t)(b * (4 * C_) + 4 * ch) * (size_t)(OH_ * OW_);
    for (int fh = 0; fh < 2; ++fh) {
        for (int fv = 0; fv < 2; ++fv) {
            v8f acc = {};
            #pragma unroll
            for (int s = 0; s < 10; ++s) {
                v2f a;                    // A[m][K] = fv[(4s+K) - 2m], K = 2*lh (+1)
                const int i0 = 4 * s + 2 * lh - 2 * lm;
                a.x = sFilt[fv][32 + i0];
                a.y = sFilt[fv][32 + i0 + 1];
                v2f bv;                   // B[K][n] = Mid[RB + 4s + K][n], K = j + 2*lh
                const int rr = RB + 4 * s + 2 * lh;
                bv.x = sMid[fh][rr][OWq + lm];
                bv.y = sMid[fh][rr + 1][OWq + lm];
                acc = __builtin_amdgcn_wmma_f32_16x16x4_f32(
                        false, a, false, bv, (short)0, acc, false, false);
            }
            const int sband = 2 * fh + fv;        // [cA, cH, cV, cD]
            const int ow = OW0 + OWq + lm;
            if (ow < OW_) {
                #pragma unroll
                for (int v = 0; v < 8; ++v) {
                    const int oh = OH0 + 16 * qi + v + 8 * lh;
                    if (oh < OH_)         // streaming, write-once: keep out of L2's way
                        __builtin_nontemporal_store(
                            acc[v], &obase[((size_t)sband * OH_ + oh) * OW_ + ow]);
                }
            }
        }
    }
}

extern "C" void kernel_launch(void* const* d_in, const int* in_sizes, int n_in,
                              void* d_out, int out_size, void* d_ws, size_t ws_size,
                              hipStream_t stream) {
    (void)in_sizes; (void)n_in; (void)d_ws; (void)ws_size; (void)out_size;
    const float* x = (const float*)d_in[0];
    float* out = (float*)d_out;
    dim3 grid(NTW * NTW, C_, B_);
    dim3 block(128);
    dwt2_db4_wmma<<<grid, block, 0, stream>>>(x, out);
}